// mLSTM_6133213299380
// MI455X (gfx1250) — compile-verified
//
#include <hip/hip_runtime.h>
#include <math.h>

// ---- sizes (fixed by the reference) ----
#define NTOK   512
#define DIN    1024
#define H      8
#define DHEAD  128
#define DOUT   1024          // H * DHEAD
#define NQKVO  4096          // 4 * DOUT
#define C_ELEMS (H * NTOK * DHEAD * DHEAD)   // 67,108,864

typedef __attribute__((ext_vector_type(2))) float v2f;
typedef __attribute__((ext_vector_type(4))) float v4f;
typedef __attribute__((ext_vector_type(8))) float v8f;

// -------------------------------------------------------------------------
// Kernel A: qkvo = x @ Wqkvo + b, with k columns pre-scaled by 1/sqrt(128).
// One wave (32 lanes) computes one 16x16 tile via V_WMMA_F32_16X16X4_F32,
// streaming K=1024 in steps of 4.  Tiles: 32 (M) x 256 (N) = 8192.
// 8 waves / 256-thread block -> 1024 blocks.
// -------------------------------------------------------------------------
__global__ __launch_bounds__(256) void gemm_qkvo_kernel(
    const float* __restrict__ x,     // (512, 1024)
    const float* __restrict__ W,     // (1024, 4096)
    const float* __restrict__ bias,  // (4096,)
    float* __restrict__ out)         // (512, 4096)
{
    const int wave = threadIdx.x >> 5;
    const int lane = threadIdx.x & 31;
    const int tile = blockIdx.x * 8 + wave;      // 0..8191
    const int mt   = tile >> 8;                  // 0..31
    const int nt   = tile & 255;                 // 0..255
    const int half = lane >> 4;                  // 0/1 -> K pair select
    const int l16  = lane & 15;

    const int row = mt * 16 + l16;               // A: M = lane%16
    const int col = nt * 16 + l16;               // B: N = lane%16

    const float* __restrict__ xa = x + (size_t)row * DIN + half * 2;
    const float* __restrict__ wb = W + (size_t)col + (size_t)(half * 2) * NQKVO;

    v8f acc = {};
#pragma unroll 8
    for (int k0 = 0; k0 < DIN; k0 += 4) {
        // A fragment: x[row][k0 + half*2 + {0,1}]  (8B-aligned pair)
        v2f a = *(const v2f*)(xa + k0);
        // B fragment: W[k0 + half*2 + {0,1}][col]
        v2f b;
        b.x = wb[(size_t)k0 * NQKVO];
        b.y = wb[(size_t)(k0 + 1) * NQKVO];
        acc = __builtin_amdgcn_wmma_f32_16x16x4_f32(
            /*neg_a=*/false, a, /*neg_b=*/false, b,
            /*c_mod=*/(short)0, acc, /*reuse_a=*/false, /*reuse_b=*/false);
    }

    // Epilogue: bias add + k-scale, D layout: row = (lane/16)*8 + i, col = lane%16
    const int comp = (col & 511) >> 7;           // 0=q 1=k 2=v 3=o within head chunk
    const float kscale = (comp == 1) ? 0.08838834764831845f /* 1/sqrt(128) */ : 1.0f;
    const float bv = bias[col];
#pragma unroll
    for (int i = 0; i < 8; ++i) {
        const int orow = mt * 16 + half * 8 + i;
        out[(size_t)orow * NQKVO + col] = (acc[i] + bv) * kscale;
    }
}

// -------------------------------------------------------------------------
// Kernel B: gates.  uf = x @ Wuf + buf  (512 x 16), then per (h, n):
//   m  = max(f + m_prev, u);  ue = exp(u - m);  fe = exp(f + m_prev - m)
// Tiny workload (16 MFLOP) -> one thread per (h, n).
// -------------------------------------------------------------------------
__global__ __launch_bounds__(256) void gates_kernel(
    const float* __restrict__ x,       // (512, 1024)
    const float* __restrict__ Wuf,     // (1024, 16)
    const float* __restrict__ buf,     // (16,)
    const float* __restrict__ m_prev,  // (8, 512, 1)
    float* __restrict__ fe_out,        // (8*512)
    float* __restrict__ ue_out)        // (8*512)
{
    const int idx = blockIdx.x * blockDim.x + threadIdx.x;   // 0..4095
    if (idx >= H * NTOK) return;
    const int h = idx >> 9;
    const int n = idx & 511;

    float u = buf[2 * h];
    float f = buf[2 * h + 1];
    const float* __restrict__ xr = x + (size_t)n * DIN;
    for (int k = 0; k < DIN; ++k) {
        const float xv = xr[k];
        u = fmaf(xv, Wuf[k * 16 + 2 * h], u);
        f = fmaf(xv, Wuf[k * 16 + 2 * h + 1], f);
    }
    const float mp = m_prev[idx];
    const float m  = fmaxf(f + mp, u);
    fe_out[idx] = expf(f + mp - m);
    ue_out[idx] = expf(u - m);
}

// -------------------------------------------------------------------------
// Kernel C: per (h, n) block (4096 blocks, 256 threads):
//   c_new = fe*c + ue*(v (outer) k)        -> streamed, non-temporal
//   kq = k.q ; npq = n_prev.q ; nm = |fe*npq + ue*kq|
//   head_out[d] = v[d]*kq / max(nm,1) * sigmoid(o[d])
// -------------------------------------------------------------------------
__global__ __launch_bounds__(256) void update_kernel(
    const float* __restrict__ c,       // (8, 512, 128, 128)
    const float* __restrict__ qkvo,    // (512, 4096)
    const float* __restrict__ n_prev,  // (8, 512, 128)
    const float* __restrict__ fe_arr,  // (8*512)
    const float* __restrict__ ue_arr,  // (8*512)
    float* __restrict__ c_new,         // (8, 512, 128, 128)  -> d_out head
    float* __restrict__ attn)          // (512, 1024) token-major head outputs
{
    const int hn = blockIdx.x;          // 0..4095
    const int h  = hn >> 9;
    const int n  = hn & 511;
    const int t  = threadIdx.x;

    __shared__ float sq[DHEAD], sk[DHEAD], sv[DHEAD], so[DHEAD];
    __shared__ float redA[256], redB[256];

    const float* __restrict__ base = qkvo + (size_t)n * NQKVO + h * 512;
    if (t < DHEAD) {
        sq[t] = base[t];
        sk[t] = base[128 + t];
        sv[t] = base[256 + t];
        so[t] = base[384 + t];
    }
    __syncthreads();

    float pk = 0.f, pn = 0.f;
    if (t < DHEAD) {
        const float qv = sq[t];
        pk = sk[t] * qv;
        pn = n_prev[(size_t)hn * DHEAD + t] * qv;
    }
    redA[t] = pk;
    redB[t] = pn;
    __syncthreads();
    for (int s = 128; s > 0; s >>= 1) {
        if (t < s) { redA[t] += redA[t + s]; redB[t] += redB[t + s]; }
        __syncthreads();
    }
    const float kq  = redA[0];
    const float npq = redB[0];

    const float fe = fe_arr[hn];
    const float ue = ue_arr[hn];
    const float nm = fabsf(fe * npq + ue * kq);
    const float denom = fmaxf(nm, 1.0f);

    if (t < DHEAD) {
        float hv = sv[t] * kq / denom;
        hv *= 1.0f / (1.0f + expf(-so[t]));
        attn[(size_t)n * DOUT + h * DHEAD + t] = hv;
    }

    // streaming c update: 16384 floats = 4096 float4, 16 per thread
    const v4f* __restrict__ cin = (const v4f*)(c + (size_t)hn * (DHEAD * DHEAD));
    v4f* __restrict__ cout      = (v4f*)(c_new + (size_t)hn * (DHEAD * DHEAD));
#pragma unroll 4
    for (int i = t; i < 4096; i += 256) {
        const int krow = i >> 5;           // /32 float4 per row
        const int l0   = (i & 31) * 4;
        const float vk = sv[krow] * ue;
        v4f cv = __builtin_nontemporal_load(cin + i);
        v4f r;
        r.x = fmaf(fe, cv.x, vk * sk[l0 + 0]);
        r.y = fmaf(fe, cv.y, vk * sk[l0 + 1]);
        r.z = fmaf(fe, cv.z, vk * sk[l0 + 2]);
        r.w = fmaf(fe, cv.w, vk * sk[l0 + 3]);
        __builtin_nontemporal_store(r, cout + i);
    }
}

// -------------------------------------------------------------------------
// Kernel D: h = LayerNorm(x + attn) per token (512 blocks, 256 threads).
// -------------------------------------------------------------------------
__global__ __launch_bounds__(256) void ln_kernel(
    const float* __restrict__ x,
    const float* __restrict__ attn,
    float* __restrict__ out)            // d_out tail (512, 1024)
{
    const int n = blockIdx.x;
    const int t = threadIdx.x;
    __shared__ float red[256];

    float vals[4];
    float s = 0.f;
#pragma unroll
    for (int i = 0; i < 4; ++i) {
        const int idx = t + i * 256;
        const float v = x[(size_t)n * DOUT + idx] + attn[(size_t)n * DOUT + idx];
        vals[i] = v;
        s += v;
    }
    red[t] = s;
    __syncthreads();
    for (int st = 128; st > 0; st >>= 1) {
        if (t < st) red[t] += red[t + st];
        __syncthreads();
    }
    const float mu = red[0] * (1.0f / DOUT);
    __syncthreads();

    float s2 = 0.f;
#pragma unroll
    for (int i = 0; i < 4; ++i) {
        const float d = vals[i] - mu;
        s2 += d * d;
    }
    red[t] = s2;
    __syncthreads();
    for (int st = 128; st > 0; st >>= 1) {
        if (t < st) red[t] += red[t + st];
        __syncthreads();
    }
    const float inv = rsqrtf(red[0] * (1.0f / DOUT) + 1e-5f);
#pragma unroll
    for (int i = 0; i < 4; ++i)
        out[(size_t)n * DOUT + t + i * 256] = (vals[i] - mu) * inv;
}

// -------------------------------------------------------------------------
extern "C" void kernel_launch(void* const* d_in, const int* in_sizes, int n_in,
                              void* d_out, int out_size, void* d_ws, size_t ws_size,
                              hipStream_t stream) {
    const float* x      = (const float*)d_in[0];   // (512, 1024)
    const float* c      = (const float*)d_in[1];   // (8, 512, 128, 128)
    const float* m_prev = (const float*)d_in[2];   // (8, 512, 1)
    const float* n_prev = (const float*)d_in[3];   // (8, 512, 128)
    const float* Wqkvo  = (const float*)d_in[4];   // (1024, 4096)
    const float* bqkvo  = (const float*)d_in[5];   // (4096,)
    const float* Wuf    = (const float*)d_in[6];   // (1024, 16)
    const float* buf    = (const float*)d_in[7];   // (16,)

    float* out_c = (float*)d_out;                  // c_new, 67,108,864 floats
    float* out_h = (float*)d_out + C_ELEMS;        // h, 524,288 floats

    // workspace layout (floats)
    float* ws    = (float*)d_ws;
    float* qkvo  = ws;                               // 512*4096 = 2,097,152
    float* fe    = qkvo + (size_t)NTOK * NQKVO;      // 4096
    float* ue    = fe + H * NTOK;                    // 4096
    float* attn  = ue + H * NTOK;                    // 512*1024 = 524,288

    gemm_qkvo_kernel<<<1024, 256, 0, stream>>>(x, Wqkvo, bqkvo, qkvo);
    gates_kernel<<<(H * NTOK + 255) / 256, 256, 0, stream>>>(x, Wuf, buf, m_prev, fe, ue);
    update_kernel<<<H * NTOK, 256, 0, stream>>>(c, qkvo, n_prev, fe, ue, out_c, attn);
    ln_kernel<<<NTOK, 256, 0, stream>>>(x, attn, out_h);
}